// ODERNNEncoder_16862041604229
// MI455X (gfx1250) — compile-verified
//
#include <hip/hip_runtime.h>
#include <hip/hip_bf16.h>

// ODE-RNN encoder for MI455X (gfx1250, wave32).
// fp32 GEMMs via V_WMMA_F32_16X16X4_F32 (reference-faithful precision).
// Decomposition: block = 4 waves cooperating on one 16-row batch group; wave w
// computes n-tile w of every 64-wide GEMM and gate tiles {w,w+4,w+8} of the
// 192-wide GRU GEMMs. 128 blocks x 4 waves = 512 waves. Activation exchange
// through shared LDS tiles uses SPLIT barriers (s_barrier_signal/-1 ... wait)
// with the next GEMM's weight-fragment ds_loads issued in the barrier shadow.
// Weights are [n][k]-major in LDS so every B fragment is one ds_load_b64
// (pad stride 66 -> conflict-free banks).

typedef __attribute__((ext_vector_type(2))) float v2f;
typedef __attribute__((ext_vector_type(8))) float v8f;

#define TSTEPS 512
#define BATCH  2048
#define OBS    16
#define HID    64
#define GATES  192
#define LSTR   66            // padded LDS row stride (dwords)
#define COOP   4             // waves per block (one 16-row group)

static constexpr int SH_FLOATS =
    HID * 18 +               // W_oh^T  [n][k], k-stride 18
    HID * LSTR +             // W1^T    [n][k] (k<64)
    HID +                    // W1 row 64 (the "t" feature row)
    HID * LSTR +             // W2^T
    GATES * LSTR +           // W_ih    [gate][k] (natural layout)
    GATES * LSTR +           // W_hh
    HID * 3 +                // b1, b2, b_oh
    GATES * 2 +              // b_ih, b_hh
    3 * 16 * LSTR;           // shared staging tiles: bufA, bufZ, bufX

// Split workgroup barrier: release LDS stores, signal, (caller does weight
// loads here), wait, acquire. Falls back to a full barrier if unavailable.
#if __has_builtin(__builtin_amdgcn_s_barrier_signal) && \
    __has_builtin(__builtin_amdgcn_s_barrier_wait)
__device__ __forceinline__ void bar_signal() {
  __builtin_amdgcn_fence(__ATOMIC_RELEASE, "workgroup");
  __builtin_amdgcn_s_barrier_signal(-1);
}
__device__ __forceinline__ void bar_wait() {
  __builtin_amdgcn_s_barrier_wait((short)-1);
  __builtin_amdgcn_fence(__ATOMIC_ACQUIRE, "workgroup");
}
#else
__device__ __forceinline__ void bar_signal() { __syncthreads(); }
__device__ __forceinline__ void bar_wait() {}
#endif

__device__ __forceinline__ v8f wmma4(v2f a, v2f b, v8f c) {
  // D = A(16x4,f32) x B(4x16,f32) + C(16x16,f32)
  return __builtin_amdgcn_wmma_f32_16x16x4_f32(
      /*neg_a=*/false, a, /*neg_b=*/false, b,
      /*c_mod=*/(short)0, c, /*reuse_a=*/false, /*reuse_b=*/false);
}

__device__ __forceinline__ float ftanh(float x) {
  float e = __expf(2.0f * x);
  return 1.0f - 2.0f / (e + 1.0f);   // stable at both extremes
}
__device__ __forceinline__ float fsigm(float x) {
  return 1.0f / (1.0f + __expf(-x));
}

// A-matrix fragments for one (16 x 64) operand: 16 K-chunks.
// Chunk kc: lanes 0-15 hold {A[m][4kc],A[m][4kc+1]}, lanes 16-31 {+2,+3}.
struct AFrag {
  v2f a[16];
};

__device__ __forceinline__ AFrag load_afrag(const float* buf, int ln, int hi) {
  AFrag f;
#pragma unroll
  for (int kc = 0; kc < 16; ++kc)
    f.a[kc] = *(const v2f*)&buf[ln * LSTR + kc * 4 + hi * 2];
  return f;
}

// B-matrix fragments for one (64 x 16) weight tile from [n][k]-major LDS.
__device__ __forceinline__ void load_bfrag(v2f* bw, const float* wT, int S,
                                           int ncol, int hi) {
#pragma unroll
  for (int kc = 0; kc < 16; ++kc)
    bw[kc] = *(const v2f*)&wT[ncol * S + kc * 4 + hi * 2];
}

// One 16x16 tile of (16x64)@(64x16): 16 chained WMMAs, B preloaded.
__device__ __forceinline__ v8f gemm_tile_pre(const AFrag& f, const v2f* bw,
                                             float bv) {
  v8f c = {bv, bv, bv, bv, bv, bv, bv, bv};
#pragma unroll
  for (int kc = 0; kc < 16; ++kc) c = wmma4(f.a[kc], bw[kc], c);
  return c;
}

// Same but with inline B loads (used where preloading isn't worth registers).
__device__ __forceinline__ v8f gemm_tile(const AFrag& f, const float* wT,
                                         int S, int ncol, float bv, int hi) {
  v8f c = {bv, bv, bv, bv, bv, bv, bv, bv};
#pragma unroll
  for (int kc = 0; kc < 16; ++kc) {
    v2f b = *(const v2f*)&wT[ncol * S + kc * 4 + hi * 2];
    c = wmma4(f.a[kc], b, c);
  }
  return c;
}

// Store one C-layout tile into a padded LDS tile at column block nt.
__device__ __forceinline__ void store_tile(float* buf, v8f v, int nt, int ln,
                                           int hi) {
#pragma unroll
  for (int i = 0; i < 8; ++i)
    buf[(i + 8 * hi) * LSTR + nt * 16 + ln] = v[i];
}

__global__ __launch_bounds__(COOP * 32) void odernn_kernel(
    const float* __restrict__ xs, const float* __restrict__ W_oh,
    const float* __restrict__ b_oh, const float* __restrict__ W1,
    const float* __restrict__ b1, const float* __restrict__ W2,
    const float* __restrict__ b2, const float* __restrict__ W_ih,
    const float* __restrict__ W_hh, const float* __restrict__ b_ih,
    const float* __restrict__ b_hh, float* __restrict__ out) {
  extern __shared__ float smem[];
  float* sWohT = smem;                       // 64*18
  float* sW1T  = sWohT + HID * 18;           // 64*66
  float* sTrow = sW1T + HID * LSTR;          // 64
  float* sW2T  = sTrow + HID;                // 64*66
  float* sWih  = sW2T + HID * LSTR;          // 192*66
  float* sWhh  = sWih + GATES * LSTR;        // 192*66
  float* sB1   = sWhh + GATES * LSTR;        // 64
  float* sB2   = sB1 + HID;                  // 64
  float* sBoh  = sB2 + HID;                  // 64
  float* sBih  = sBoh + HID;                 // 192
  float* sBhh  = sBih + GATES;               // 192
  float* bufA  = sBhh + GATES;               // 16*66 (odef input h_arg / h)
  float* bufZ  = bufA + 16 * LSTR;           // 16*66 (tanh intermediate)
  float* bufX  = bufZ + 16 * LSTR;           // 16*66 (obs2hidden result)

  const int tid = threadIdx.x;
  const int nth = blockDim.x;

  // ---- one-time cooperative weight preload (transpose to [n][k]) ----
  for (int i = tid; i < HID * OBS; i += nth) {
    int n = i >> 4, k = i & 15;
    sWohT[n * 18 + k] = W_oh[k * HID + n];
  }
  for (int i = tid; i < HID * HID; i += nth) {
    int n = i >> 6, k = i & 63;
    sW1T[n * LSTR + k] = W1[k * HID + n];
    sW2T[n * LSTR + k] = W2[k * HID + n];
  }
  for (int i = tid; i < GATES * HID; i += nth) {
    int g = i >> 6, k = i & 63;
    sWih[g * LSTR + k] = W_ih[i];   // already [gate][k]
    sWhh[g * LSTR + k] = W_hh[i];
  }
  for (int i = tid; i < HID; i += nth) {
    sTrow[i] = W1[HID * HID + i];   // W1 row 64 (t feature)
    sB1[i] = b1[i];
    sB2[i] = b2[i];
    sBoh[i] = b_oh[i];
  }
  for (int i = tid; i < GATES; i += nth) {
    sBih[i] = b_ih[i];
    sBhh[i] = b_hh[i];
  }
  __syncthreads();

  const int w = tid >> 5;          // wave id = owned n-tile
  const int lane = tid & 31;
  const int ln = lane & 15;
  const int hi = lane >> 4;
  const int ncol = w * 16 + ln;    // this lane's n within owned HID tile
  const int rowBase = blockIdx.x * 16;

  const float dt = 1.0f / 4.0f;
  const v8f vzero = {0.f, 0.f, 0.f, 0.f, 0.f, 0.f, 0.f, 0.f};

  v8f h = vzero;                   // this wave's 16x16 slice of hidden state

  for (int t = 0; t < TSTEPS; ++t) {
    // ---- prefetch next timestep's x tile (only HBM traffic in the loop) ----
    if (t + 1 < TSTEPS) {
      const float* xnext = xs + ((size_t)(t + 1) * BATCH + rowBase) * OBS;
      __builtin_prefetch(xnext + lane * 8, 0, 3);  // 32 lanes x 32B = 1KB tile
    }

    // ---- obs2hidden: leaky_relu(x_t @ W_oh + b_oh); wave w -> tile w ----
    {
      const float* xrow = xs + ((size_t)t * BATCH + rowBase) * OBS;
      float bv = sBoh[ncol];
      v8f c = {bv, bv, bv, bv, bv, bv, bv, bv};
#pragma unroll
      for (int kc = 0; kc < 4; ++kc) {
        v2f xa = *(const v2f*)&xrow[ln * OBS + kc * 4 + hi * 2];
        v2f b = *(const v2f*)&sWohT[ncol * 18 + kc * 4 + hi * 2];
        c = wmma4(xa, b, c);
      }
#pragma unroll
      for (int i = 0; i < 8; ++i) {
        float v = c[i];
        c[i] = fmaxf(v, 0.f) + 0.1f * fminf(v, 0.f);  // leaky_relu(0.1)
      }
      store_tile(bufX, c, w, ln, hi);
      // bufX is read only in the GRU phase, which is separated from this
      // store by the RK4 barriers below -> no extra barrier needed here.
    }

    // ---- RK4 over t_ode in [0,1], 4 fixed steps ----
    float tt = 0.f;
    v8f kk = vzero;
#pragma unroll 1
    for (int s = 0; s < 4; ++s) {
      v8f ksum = vzero;
#pragma unroll 1
      for (int sub = 0; sub < 4; ++sub) {
        // input coefficient on previous k: 0, dt/2, dt/2, dt (== t offset)
        float ci = (sub == 0) ? 0.f : ((sub == 3) ? dt : 0.5f * dt);
        float wk = (sub == 1 || sub == 2) ? 2.f : 1.f;
        v8f harg;
#pragma unroll
        for (int i = 0; i < 8; ++i) harg[i] = h[i] + ci * kk[i];
        store_tile(bufA, harg, w, ln, hi);
        bar_signal();                          // B1: publish bufA
        // barrier shadow: W1 fragments + effective bias (independent of bufA)
        v2f bw[16];
        load_bfrag(bw, sW1T, LSTR, ncol, hi);
        float tv = tt + ci;
        float bv1 = sB1[ncol] + tv * sTrow[ncol];
        bar_wait();
        // odef layer 1: z = tanh([h,t] @ W1 + b1) -- this wave's n-tile
        AFrag af = load_afrag(bufA, ln, hi);
        v8f z = gemm_tile_pre(af, bw, bv1);
#pragma unroll
        for (int i = 0; i < 8; ++i) z[i] = ftanh(z[i]);
        store_tile(bufZ, z, w, ln, hi);
        bar_signal();                          // B2: publish bufZ
        // barrier shadow: W2 fragments
        load_bfrag(bw, sW2T, LSTR, ncol, hi);
        float bv2 = sB2[ncol];
        bar_wait();
        // odef layer 2: k = z @ W2 + b2 -- this wave's n-tile
        AFrag az = load_afrag(bufZ, ln, hi);
        kk = gemm_tile_pre(az, bw, bv2);
#pragma unroll
        for (int i = 0; i < 8; ++i) ksum[i] += wk * kk[i];
      }
#pragma unroll
      for (int i = 0; i < 8; ++i) h[i] += (dt / 6.0f) * ksum[i];
      tt += dt;
    }

    // ---- GRU cell: gates stacked [r, z, n]; wave w -> tiles w, w+4, w+8 ----
    store_tile(bufA, h, w, ln, hi);  // h_ode as shared A source
    bar_signal();                              // B_gru: publish h_ode
    // barrier shadow: first gate tile's weight fragments
    v2f bwh[16], bwx[16];
    load_bfrag(bwh, sWhh, LSTR, w * 16 + ln, hi);
    load_bfrag(bwx, sWih, LSTR, w * 16 + ln, hi);
    bar_wait();
    {
      AFrag afH = load_afrag(bufA, ln, hi);
      AFrag afX = load_afrag(bufX, ln, hi);
      v8f gh[3], gx[3];
      gh[0] = gemm_tile_pre(afH, bwh, sBhh[ncol]);
      gx[0] = gemm_tile_pre(afX, bwx, sBih[ncol]);
#pragma unroll
      for (int j = 1; j < 3; ++j) {
        int gcol = (w + 4 * j) * 16 + ln;              // z, n slices
        gh[j] = gemm_tile(afH, sWhh, LSTR, gcol, sBhh[gcol], hi);
        gx[j] = gemm_tile(afX, sWih, LSTR, gcol, sBih[gcol], hi);
      }
#pragma unroll
      for (int i = 0; i < 8; ++i) {
        float r = fsigm(gx[0][i] + gh[0][i]);
        float zz = fsigm(gx[1][i] + gh[1][i]);
        float nn = ftanh(gx[2][i] + r * gh[2][i]);
        h[i] = (1.f - zz) * nn + zz * h[i];
      }
    }
    bar_signal();  // B_end: protect bufA/bufX against next-step overwrite
    bar_wait();
  }

  // ---- write h_final (B, HID): wave w writes its 16-column slice ----
#pragma unroll
  for (int i = 0; i < 8; ++i)
    out[(size_t)(rowBase + i + 8 * hi) * HID + ncol] = h[i];
}

extern "C" void kernel_launch(void* const* d_in, const int* in_sizes, int n_in,
                              void* d_out, int out_size, void* d_ws,
                              size_t ws_size, hipStream_t stream) {
  const float* xs   = (const float*)d_in[0];
  const float* W_oh = (const float*)d_in[1];
  const float* b_oh = (const float*)d_in[2];
  const float* W1   = (const float*)d_in[3];
  const float* b1   = (const float*)d_in[4];
  const float* W2   = (const float*)d_in[5];
  const float* b2   = (const float*)d_in[6];
  const float* W_ih = (const float*)d_in[7];
  const float* W_hh = (const float*)d_in[8];
  const float* b_ih = (const float*)d_in[9];
  const float* b_hh = (const float*)d_in[10];
  float* out = (float*)d_out;

  dim3 grid(BATCH / 16);             // 128 blocks, one 16-row group each
  dim3 block(COOP * 32);             // 4 cooperating waves
  size_t shbytes = (size_t)SH_FLOATS * sizeof(float);  // ~155 KB
  odernn_kernel<<<grid, block, shbytes, stream>>>(
      xs, W_oh, b_oh, W1, b1, W2, b2, W_ih, W_hh, b_ih, b_hh, out);
}